// GINNet_7859790152295
// MI455X (gfx1250) — compile-verified
//
#include <hip/hip_runtime.h>
#include <math.h>

#define NNODES 100000
#define NEDGES 1600000
#define DIM 32
#define FIN 128
#define NCLS 40
#define BN_EPS 1e-5f

typedef __attribute__((ext_vector_type(2))) float v2f;
typedef __attribute__((ext_vector_type(8))) float v8f;

// D = A(16x4) * B(4x16) + C, fp32 WMMA (CDNA5 V_WMMA_F32_16X16X4_F32)
__device__ __forceinline__ v8f wmma4(v2f a, v2f b, v8f c) {
    return __builtin_amdgcn_wmma_f32_16x16x4_f32(
        /*neg_a=*/false, a, /*neg_b=*/false, b,
        /*c_mod=*/(short)0, c, /*reuse_a=*/false, /*reuse_b=*/false);
}

// ---------------------------------------------------------------------------
// Kernel 1: Y(N x 32) = X(N x 128) @ W(128 x 32), fp32 WMMA, W staged in LDS.
// One wave computes a 16x32 output block (two 16x16 accumulators), K swept in
// steps of 4 (32 wmma pairs).
// ---------------------------------------------------------------------------
__global__ void __launch_bounds__(256)
gemm_fin_kernel(const float* __restrict__ X, const float* __restrict__ W,
                float* __restrict__ Y, int nrows) {
    __shared__ float lW[FIN * DIM];  // 16 KB
    for (int i = threadIdx.x; i < FIN * DIM; i += blockDim.x) lW[i] = W[i];
    __syncthreads();

    int lane = threadIdx.x & 31;
    int m0 = (blockIdx.x * (blockDim.x >> 5) + (threadIdx.x >> 5)) * 16;
    if (m0 >= nrows) return;  // wave-uniform exit: EXEC stays all-ones for WMMA

    int half = lane >> 4;    // 0: lanes 0-15, 1: lanes 16-31
    int lm   = lane & 15;
    int row  = m0 + lm; if (row >= nrows) row = nrows - 1;
    const float* xr = X + (size_t)row * FIN;

    v8f c0 = {}; v8f c1 = {};
#pragma unroll 8
    for (int kk = 0; kk < FIN / 4; ++kk) {
        int k = kk * 4 + 2 * half;
        v2f a, b0, b1;
        a.x  = xr[k];                 a.y  = xr[k + 1];                 // contiguous 8B
        b0.x = lW[k * DIM + lm];      b0.y = lW[(k + 1) * DIM + lm];
        b1.x = lW[k * DIM + 16 + lm]; b1.y = lW[(k + 1) * DIM + 16 + lm];
        c0 = wmma4(a, b0, c0);
        c1 = wmma4(a, b1, c1);
    }
#pragma unroll
    for (int r = 0; r < 8; ++r) {
        int orow = m0 + r + 8 * half;
        if (orow < nrows) {
            Y[(size_t)orow * DIM + lm]      = c0[r];
            Y[(size_t)orow * DIM + 16 + lm] = c1[r];
        }
    }
}

// ---------------------------------------------------------------------------
// Kernel 2: zero a float buffer (agg must be zeroed on every launch).
// ---------------------------------------------------------------------------
__global__ void __launch_bounds__(256)
zero_kernel(float4* __restrict__ p, int n4) {
    int i = blockIdx.x * blockDim.x + threadIdx.x;
    if (i < n4) p[i] = make_float4(0.f, 0.f, 0.f, 0.f);
}

// ---------------------------------------------------------------------------
// Kernel 3: edge scatter-add: agg[dst] += y[src].  8 threads per edge, each
// handles one float4 chunk of the 32-dim feature.  The 12.8MB tables are L2
// resident, so both the gather and the f32 atomics stay in L2.
// ---------------------------------------------------------------------------
__global__ void __launch_bounds__(256)
scatter_kernel(const float* __restrict__ y, const int* __restrict__ src,
               const int* __restrict__ dst, float* __restrict__ agg, int nedges) {
    int tid = blockIdx.x * blockDim.x + threadIdx.x;
    int e  = tid >> 3;
    int c4 = (tid & 7) * 4;
    if (e >= nedges) return;
    int s = src[e], d = dst[e];
    float4 v = *(const float4*)(y + (size_t)s * DIM + c4);
    float* p = agg + (size_t)d * DIM + c4;
    atomicAdd(p + 0, v.x);
    atomicAdd(p + 1, v.y);
    atomicAdd(p + 2, v.z);
    atomicAdd(p + 3, v.w);
}

// ---------------------------------------------------------------------------
// Kernel 4: fused 32x32 GEMM + GIN epilogues, fp32 WMMA, weights in LDS.
//   MODE 0: Y = A0 @ W                                     (plain projection)
//   MODE 1: t = relu(A0 + A1 + preb); Y = BN(relu(t@W + postb))   (conv1)
//   MODE 2: t = relu(A0 + A1 + preb); Y = BN(t@W + postb)         (conv2)
// ---------------------------------------------------------------------------
template <int MODE>
__global__ void __launch_bounds__(256)
gemm32_kernel(const float* __restrict__ A0, const float* __restrict__ A1,
              const float* __restrict__ W,  const float* __restrict__ preb,
              const float* __restrict__ postb,
              const float* __restrict__ bng, const float* __restrict__ bnb,
              const float* __restrict__ bnm, const float* __restrict__ bnv,
              float* __restrict__ Y, int nrows) {
    __shared__ float lW[DIM * DIM];
    __shared__ float lpre[DIM], lpost[DIM], lscale[DIM], lshift[DIM];
    for (int i = threadIdx.x; i < DIM * DIM; i += blockDim.x) lW[i] = W[i];
    if (MODE != 0 && threadIdx.x < DIM) {
        int c = threadIdx.x;
        lpre[c]  = preb[c];
        lpost[c] = postb[c];
        float sc = bng[c] * rsqrtf(bnv[c] + BN_EPS);
        lscale[c] = sc;
        lshift[c] = bnb[c] - bnm[c] * sc;
    }
    __syncthreads();

    int lane = threadIdx.x & 31;
    int m0 = (blockIdx.x * (blockDim.x >> 5) + (threadIdx.x >> 5)) * 16;
    if (m0 >= nrows) return;  // wave-uniform

    int half = lane >> 4;
    int lm   = lane & 15;
    int row  = m0 + lm; if (row >= nrows) row = nrows - 1;
    const float* ar0 = A0 + (size_t)row * DIM;
    const float* ar1 = (MODE != 0) ? (A1 + (size_t)row * DIM) : nullptr;

    v8f c0 = {}; v8f c1 = {};
#pragma unroll
    for (int kk = 0; kk < DIM / 4; ++kk) {
        int k = kk * 4 + 2 * half;
        v2f a, b0, b1;
        if (MODE == 0) {
            a.x = ar0[k];
            a.y = ar0[k + 1];
        } else {
            a.x = fmaxf(ar0[k]     + ar1[k]     + lpre[k],     0.f);
            a.y = fmaxf(ar0[k + 1] + ar1[k + 1] + lpre[k + 1], 0.f);
        }
        b0.x = lW[k * DIM + lm];      b0.y = lW[(k + 1) * DIM + lm];
        b1.x = lW[k * DIM + 16 + lm]; b1.y = lW[(k + 1) * DIM + 16 + lm];
        c0 = wmma4(a, b0, c0);
        c1 = wmma4(a, b1, c1);
    }
#pragma unroll
    for (int r = 0; r < 8; ++r) {
        int orow = m0 + r + 8 * half;
        if (orow >= nrows) continue;
        float u0 = c0[r], u1 = c1[r];
        if (MODE == 0) {
            Y[(size_t)orow * DIM + lm]      = u0;
            Y[(size_t)orow * DIM + 16 + lm] = u1;
        } else {
            float h0 = u0 + lpost[lm];
            float h1 = u1 + lpost[16 + lm];
            if (MODE == 1) { h0 = fmaxf(h0, 0.f); h1 = fmaxf(h1, 0.f); }
            Y[(size_t)orow * DIM + lm]      = h0 * lscale[lm]      + lshift[lm];
            Y[(size_t)orow * DIM + 16 + lm] = h1 * lscale[16 + lm] + lshift[16 + lm];
        }
    }
}

// ---------------------------------------------------------------------------
// Kernel 5: classifier head, one thread per node.
//   h3 = relu(h2 @ fc1 + b1); logits = h3 @ fc2 + b2; out = log_softmax
// ---------------------------------------------------------------------------
__global__ void __launch_bounds__(256)
head_kernel(const float* __restrict__ H,
            const float* __restrict__ fc1w, const float* __restrict__ fc1b,
            const float* __restrict__ fc2w, const float* __restrict__ fc2b,
            float* __restrict__ out, int nrows) {
    __shared__ float w1[DIM * DIM], b1[DIM], w2[DIM * NCLS], b2[NCLS];
    for (int i = threadIdx.x; i < DIM * DIM;  i += blockDim.x) w1[i] = fc1w[i];
    for (int i = threadIdx.x; i < DIM * NCLS; i += blockDim.x) w2[i] = fc2w[i];
    if (threadIdx.x < DIM)  b1[threadIdx.x] = fc1b[threadIdx.x];
    if (threadIdx.x < NCLS) b2[threadIdx.x] = fc2b[threadIdx.x];
    __syncthreads();

    int i = blockIdx.x * blockDim.x + threadIdx.x;
    if (i >= nrows) return;

    float h[DIM];
    const float4* hp = (const float4*)(H + (size_t)i * DIM);
#pragma unroll
    for (int q = 0; q < DIM / 4; ++q) {
        float4 v = hp[q];
        h[q * 4 + 0] = v.x; h[q * 4 + 1] = v.y; h[q * 4 + 2] = v.z; h[q * 4 + 3] = v.w;
    }

    float t[DIM];
#pragma unroll
    for (int c = 0; c < DIM; ++c) t[c] = b1[c];
#pragma unroll 4
    for (int k = 0; k < DIM; ++k) {
        float hk = h[k];
#pragma unroll
        for (int c = 0; c < DIM; ++c) t[c] = fmaf(hk, w1[k * DIM + c], t[c]);
    }
#pragma unroll
    for (int c = 0; c < DIM; ++c) t[c] = fmaxf(t[c], 0.f);

    float l[NCLS];
#pragma unroll
    for (int c = 0; c < NCLS; ++c) l[c] = b2[c];
#pragma unroll 4
    for (int k = 0; k < DIM; ++k) {
        float tk = t[k];
#pragma unroll
        for (int c = 0; c < NCLS; ++c) l[c] = fmaf(tk, w2[k * NCLS + c], l[c]);
    }

    float mx = l[0];
#pragma unroll
    for (int c = 1; c < NCLS; ++c) mx = fmaxf(mx, l[c]);
    float s = 0.f;
#pragma unroll
    for (int c = 0; c < NCLS; ++c) s += expf(l[c] - mx);
    float lse = mx + logf(s);

    float* op = out + (size_t)i * NCLS;
#pragma unroll
    for (int c = 0; c < NCLS; ++c) op[c] = l[c] - lse;
}

// ---------------------------------------------------------------------------
extern "C" void kernel_launch(void* const* d_in, const int* in_sizes, int n_in,
                              void* d_out, int out_size, void* d_ws, size_t ws_size,
                              hipStream_t stream) {
    (void)in_sizes; (void)n_in; (void)out_size; (void)ws_size;

    const float* x    = (const float*)d_in[0];
    const int*   ei   = (const int*)d_in[1];   // harness delivers int32 indices
    const float* w1a  = (const float*)d_in[2];
    const float* b1a  = (const float*)d_in[3];
    const float* w1b  = (const float*)d_in[4];
    const float* b1b  = (const float*)d_in[5];
    const float* w2a  = (const float*)d_in[6];
    const float* b2a  = (const float*)d_in[7];
    const float* w2b  = (const float*)d_in[8];
    const float* b2b  = (const float*)d_in[9];
    const float* bn1g = (const float*)d_in[10];
    const float* bn1b = (const float*)d_in[11];
    const float* bn1m = (const float*)d_in[12];
    const float* bn1v = (const float*)d_in[13];
    const float* bn2g = (const float*)d_in[14];
    const float* bn2b = (const float*)d_in[15];
    const float* bn2m = (const float*)d_in[16];
    const float* bn2v = (const float*)d_in[17];
    const float* fc1w = (const float*)d_in[18];
    const float* fc1b = (const float*)d_in[19];
    const float* fc2w = (const float*)d_in[20];
    const float* fc2b = (const float*)d_in[21];

    const int* src = ei;           // edge_index[0]
    const int* dst = ei + NEDGES;  // edge_index[1]

    // 3 rotating N x 32 f32 buffers (12.8 MB each, 38.4 MB total)
    float* buf0 = (float*)d_ws;
    float* buf1 = buf0 + (size_t)NNODES * DIM;
    float* buf2 = buf1 + (size_t)NNODES * DIM;

    dim3 blk(256);
    int gemmBlocks    = (NNODES + 16 * 8 - 1) / (16 * 8);            // 782
    int zeroBlocks    = (NNODES * DIM / 4 + 255) / 256;              // 3125
    int scatterBlocks = (int)(((long long)NEDGES * 8 + 255) / 256);  // 50000
    int headBlocks    = (NNODES + 255) / 256;                        // 391

    // ---- Conv1: project -> aggregate (in projected space) -> fused MLP/BN ----
    gemm_fin_kernel<<<gemmBlocks, blk, 0, stream>>>(x, w1a, buf0, NNODES);         // y1 = x@w1a
    zero_kernel<<<zeroBlocks, blk, 0, stream>>>((float4*)buf1, NNODES * DIM / 4);
    scatter_kernel<<<scatterBlocks, blk, 0, stream>>>(buf0, src, dst, buf1, NEDGES);
    gemm32_kernel<1><<<gemmBlocks, blk, 0, stream>>>(buf1, buf0, w1b, b1a, b1b,
                                                     bn1g, bn1b, bn1m, bn1v,
                                                     buf2, NNODES);                // h1

    // ---- Conv2 ----
    gemm32_kernel<0><<<gemmBlocks, blk, 0, stream>>>(buf2, nullptr, w2a, nullptr,
                                                     nullptr, nullptr, nullptr,
                                                     nullptr, nullptr,
                                                     buf0, NNODES);                // y2 = h1@w2a
    zero_kernel<<<zeroBlocks, blk, 0, stream>>>((float4*)buf1, NNODES * DIM / 4);
    scatter_kernel<<<scatterBlocks, blk, 0, stream>>>(buf0, src, dst, buf1, NEDGES);
    gemm32_kernel<2><<<gemmBlocks, blk, 0, stream>>>(buf1, buf0, w2b, b2a, b2b,
                                                     bn2g, bn2b, bn2m, bn2v,
                                                     buf2, NNODES);                // h2

    // ---- Head ----
    head_kernel<<<headBlocks, blk, 0, stream>>>(buf2, fc1w, fc1b, fc2w, fc2b,
                                                (float*)d_out, NNODES);
}